// EntityMemory_4234837753995
// MI455X (gfx1250) — compile-verified
//
#include <hip/hip_runtime.h>
#include <cfloat>
#include <cstdint>

#define ENTITY_START 32100
#define Bn 4
#define Sn 2048
#define Dn 1024
#define DEn 256
#define Vn 1000000
#define NMENT 128
#define TOPK 100
#define NT_PER_BLK 4     // 4 n-tiles of 16 columns per block in k3

typedef float v8f __attribute__((ext_vector_type(8)));
typedef float v2f __attribute__((ext_vector_type(2)));
typedef unsigned int u32x4 __attribute__((ext_vector_type(4)));
typedef int i32x4 __attribute__((ext_vector_type(4)));
typedef int i32x8 __attribute__((ext_vector_type(8)));

__device__ __forceinline__ unsigned f2key(float f) {
    unsigned u = __float_as_uint(f);
    return (u & 0x80000000u) ? ~u : (u | 0x80000000u);
}
__device__ __forceinline__ float key2f(unsigned k) {
    unsigned u = (k & 0x80000000u) ? (k ^ 0x80000000u) : ~k;
    return __uint_as_float(u);
}

// Issue a TDM DMA: copy a 2D tile (tile_dim0=16 cols x tile_dim1=256 rows, f32)
// of a row-major tensor with row stride Vn elements into LDS at lds_off.
// D# bitfields per CDNA5 ISA. One issue per wave (EXEC ignored).
__device__ __forceinline__ void tdm_load_tile16x256(unsigned lds_off, const float* gptr) {
    unsigned long long ga = (unsigned long long)(uintptr_t)gptr;
    u32x4 g0;
    g0.x = 1u;                                            // count=1, is_restore=0, gather off
    g0.y = lds_off;                                       // lds_addr (bytes)
    g0.z = (unsigned)(ga & 0xFFFFFFFFu);                  // global_addr[31:0]
    g0.w = (unsigned)((ga >> 32) & 0x01FFFFFFu)           // global_addr[56:32]
         | (2u << 30);                                    // type=2 ("image")
    i32x8 g1;
    g1[0] = (int)(2u << 16);                              // workgroup_mask=0, data_size=2 (4B)
    g1[1] = (int)(((unsigned)Vn & 0xFFFFu) << 16);        // tensor_dim0[15:0] @ bits 63:48
    g1[2] = (int)(((unsigned)Vn >> 16)                    // tensor_dim0[31:16]
         | (256u << 16));                                 // tensor_dim1[15:0]
    g1[3] = (int)(16u << 16);                             // tensor_dim1[31:16]=0 | tile_dim0=16
    g1[4] = (int)256u;                                    // tile_dim1=256, tile_dim2=0 (2D)
    g1[5] = (int)Vn;                                      // tensor_dim0_stride[31:0]
    g1[6] = 0;                                            // stride[47:32]=0 | dim1_stride lo=0
    g1[7] = 0;                                            // dim1_stride hi=0 (ignored, 2D)
    i32x4 z4 = {0, 0, 0, 0};
#if defined(__clang_major__) && (__clang_major__ >= 23)
    i32x8 z8 = {0, 0, 0, 0, 0, 0, 0, 0};
    __builtin_amdgcn_tensor_load_to_lds(g0, g1, z4, z4, z8, 0);
#else
    __builtin_amdgcn_tensor_load_to_lds(g0, g1, z4, z4, 0);
#endif
}

// ---------------- K0: zero the (B,S,D) output region ----------------
__global__ void k0_zero(float4* __restrict__ out, int n4) {
    int i = blockIdx.x * blockDim.x + threadIdx.x;
    int stride = gridDim.x * blockDim.x;
    float4 z; z.x = 0.f; z.y = 0.f; z.z = 0.f; z.w = 0.f;
    for (; i < n4; i += stride) out[i] = z;
}

// ---------------- K1: ordered nonzero scan for mention positions ----------------
__global__ void k1_find(const int* __restrict__ ids, int* __restrict__ pos) {
    __shared__ int scanb[1024];
    int t = threadIdx.x;
    if (t < NMENT) pos[t] = 0;
    int base = t * 8;
    int loc[8];
    int cnt = 0;
    for (int j = 0; j < 8; j++) {
        int v = ids[base + j];
        loc[j] = (v == ENTITY_START) ? 1 : 0;
        cnt += loc[j];
    }
    scanb[t] = cnt;
    __syncthreads();
    for (int off = 1; off < 1024; off <<= 1) {
        int v = (t >= off) ? scanb[t - off] : 0;
        __syncthreads();
        scanb[t] += v;
        __syncthreads();
    }
    int run = scanb[t] - cnt;   // exclusive prefix
    for (int j = 0; j < 8; j++) {
        if (loc[j]) {
            if (run < NMENT) pos[run] = base + j;   // flat index = row*S + col
            run++;
        }
    }
}

// ---------------- K2: query = gather(hidden)[128x1024] @ W_in^T + b_in  (WMMA f32 16x16x4) ----------------
__global__ void k2_query(const float* __restrict__ hs, const int* __restrict__ pos,
                         const float* __restrict__ Win, const float* __restrict__ bin,
                         float* __restrict__ query) {
    int wave = threadIdx.x >> 5, lane = threadIdx.x & 31;
    int gw = blockIdx.x * 8 + wave;     // 128 waves total
    int mt = gw >> 4, nt = gw & 15;     // 8 m-tiles x 16 n-tiles
    int halfsel = lane >> 4;
    int l15 = lane & 15;
    int koff = 2 * halfsel;
    int arow = pos[mt * 16 + l15];
    int ncol = nt * 16 + l15;
    const float* aptr = hs + (size_t)arow * Dn;
    const float* bptr = Win + (size_t)ncol * Dn;   // B[k][n] = W_in[n*Dn + k]
    v8f c = {};
    #pragma unroll 4
    for (int k = 0; k < Dn; k += 4) {
        v2f a = *(const v2f*)(aptr + k + koff);
        v2f b = *(const v2f*)(bptr + k + koff);
        c = __builtin_amdgcn_wmma_f32_16x16x4_f32(false, a, false, b, (short)0, c, false, false);
    }
    float bb = bin[ncol];
    for (int i = 0; i < 8; i++) {
        int m = mt * 16 + i + 8 * halfsel;
        query[(size_t)m * DEn + ncol] = c[i] + bb;
    }
}

// ---------------- K3: score = query[128x256] @ E_w[256x1e6], mask, store monotonic u32 keys ----
// Block = 8 waves (one 16-row m-tile each) x 4 n-tiles. E_w tiles (16 cols x 256 rows, 16KB)
// are streamed into LDS by the Tensor Data Mover, double-buffered; WMMA reads B from LDS.
__global__ void k3_score(const float* __restrict__ query, const float* __restrict__ Ew,
                         const int* __restrict__ mask, unsigned* __restrict__ skey) {
    __shared__ float tile[2][16 * 256];   // two 16KB TDM buffers, layout [k][ncol]
    int wave = threadIdx.x >> 5, lane = threadIdx.x & 31;
    int mt = wave;
    int halfsel = lane >> 4, l15 = lane & 15;
    int koff = 2 * halfsel;
    long n0 = (long)blockIdx.x * (16 * NT_PER_BLK);
    const float* aptr = query + (size_t)(mt * 16 + l15) * DEn;
    unsigned mkey = f2key(-FLT_MAX);

    // prologue: wave 0 kicks off DMA of tile 0
    if (threadIdx.x < 32) {
        tdm_load_tile16x256((unsigned)(uintptr_t)&tile[0][0], Ew + n0);
    }

    for (int it = 0; it < NT_PER_BLK; it++) {
        if (threadIdx.x < 32) {
            if (it + 1 < NT_PER_BLK) {
                tdm_load_tile16x256((unsigned)(uintptr_t)&tile[(it + 1) & 1][0],
                                    Ew + n0 + (long)(it + 1) * 16);
                __builtin_amdgcn_s_wait_tensorcnt((short)1);  // tile 'it' done, next in flight
            } else {
                __builtin_amdgcn_s_wait_tensorcnt((short)0);
            }
        }
        __syncthreads();   // DMA-complete + prior reads of this buffer finished

        const float* bt = &tile[it & 1][0];
        v8f c = {};
        #pragma unroll 4
        for (int k = 0; k < DEn; k += 4) {
            v2f a = *(const v2f*)(aptr + k + koff);
            v2f b;
            b.x = bt[(k + koff) * 16 + l15];
            b.y = bt[(k + koff + 1) * 16 + l15];
            c = __builtin_amdgcn_wmma_f32_16x16x4_f32(false, a, false, b, (short)0, c, false, false);
        }
        long n = n0 + (long)it * 16 + l15;
        bool masked = (mask[n] != 0);
        for (int i = 0; i < 8; i++) {
            int m = mt * 16 + i + 8 * halfsel;
            unsigned key = masked ? mkey : f2key(c[i]);
            __builtin_nontemporal_store(key, skey + (size_t)m * Vn + n);
        }
        __syncthreads();   // all reads of tile[it&1] done before it is re-filled
    }
}

// ---------------- K4: exact top-100 per mention (radix select) + softmax + pick ----------------
__global__ void k4_topk(const unsigned* __restrict__ skey, const float* __restrict__ Ew,
                        float* __restrict__ pick, int* __restrict__ topi_out) {
    __shared__ unsigned hist[4096];
    __shared__ unsigned sPrefix;
    __shared__ unsigned sCntGt, sTie;
    __shared__ unsigned candKey[160];
    __shared__ int candIdx[160];
    __shared__ int tieIdx[1024];
    __shared__ unsigned finKey[TOPK + 32];
    __shared__ int finIdx[TOPK + 32];
    __shared__ float wgt[TOPK];
    __shared__ float sDen;

    int m = blockIdx.x, t = threadIdx.x, nthr = blockDim.x;
    const unsigned* base = skey + (size_t)m * Vn;

    // --- 3-pass radix select (MSB-first: 12 + 12 + 8 bits) for the 100th-largest key ---
    unsigned prefix = 0;
    unsigned need = TOPK;
    const int shifts[3] = {20, 8, 0};
    const int nbits[3]  = {12, 12, 8};
    for (int p = 0; p < 3; p++) {
        int sh = shifts[p];
        int nb = 1 << nbits[p];
        unsigned maskHi = (p == 0) ? 0u : ~((1u << (sh + nbits[p])) - 1u);
        for (int i = t; i < nb; i += nthr) hist[i] = 0;
        __syncthreads();
        for (long i = t; i < Vn; i += nthr) {
            unsigned k = base[i];
            if ((k & maskHi) == (prefix & maskHi))
                atomicAdd(&hist[(k >> sh) & (nb - 1)], 1u);
        }
        __syncthreads();
        if (t == 0) {
            unsigned cum = 0;
            int b = nb - 1;
            for (; b > 0; b--) {
                if (cum + hist[b] >= need) break;
                cum += hist[b];
            }
            prefix |= ((unsigned)b) << sh;
            need -= cum;
            sPrefix = prefix;
        }
        __syncthreads();
        prefix = sPrefix;
        __syncthreads();
    }
    unsigned T = prefix;   // 100th largest key; count(>T) < 100 <= count(>=T)

    // --- collect candidates ---
    if (t == 0) { sCntGt = 0; sTie = 0; }
    __syncthreads();
    for (long i = t; i < Vn; i += nthr) {
        unsigned k = base[i];
        if (k > T) {
            unsigned pp = atomicAdd(&sCntGt, 1u);
            if (pp < 160u) { candKey[pp] = k; candIdx[pp] = (int)i; }
        } else if (k == T) {
            unsigned pp = atomicAdd(&sTie, 1u);
            if (pp < 1024u) tieIdx[pp] = (int)i;
        }
    }
    __syncthreads();

    if (t == 0) {
        int nGt = (sCntGt < 160u) ? (int)sCntGt : 160;
        if (nGt > TOPK) nGt = TOPK;
        int tn = (sTie < 1024u) ? (int)sTie : 1024;
        int take = TOPK - nGt;
        if (take < 0) take = 0;
        if (take > tn) take = tn;
        // pick the 'take' smallest tie indices (jax tie-break: lower index first)
        for (int j = 0; j < take; j++) {
            int mb = j;
            for (int q = j + 1; q < tn; q++)
                if (tieIdx[q] < tieIdx[mb]) mb = q;
            int tv = tieIdx[mb]; tieIdx[mb] = tieIdx[j]; tieIdx[j] = tv;
        }
        int tot = 0;
        for (int i = 0; i < nGt; i++) { finKey[tot] = candKey[i]; finIdx[tot] = candIdx[i]; tot++; }
        for (int j = 0; j < take; j++) { finKey[tot] = T; finIdx[tot] = tieIdx[j]; tot++; }
        // insertion sort: key desc, index asc on ties
        for (int a = 1; a < tot; a++) {
            unsigned kk = finKey[a]; int id = finIdx[a];
            int b2 = a - 1;
            while (b2 >= 0 && (finKey[b2] < kk || (finKey[b2] == kk && finIdx[b2] > id))) {
                finKey[b2 + 1] = finKey[b2]; finIdx[b2 + 1] = finIdx[b2]; b2--;
            }
            finKey[b2 + 1] = kk; finIdx[b2 + 1] = id;
        }
        for (int i = tot; i < TOPK; i++) { finKey[i] = f2key(-FLT_MAX); finIdx[i] = 0; }
    }
    __syncthreads();

    // write topi (int32 bit-cast into the f32 output tail)
    for (int k = t; k < TOPK; k += nthr) topi_out[m * TOPK + k] = finIdx[k];

    // softmax over top values
    if (t == 0) {
        float mx = key2f(finKey[0]);
        float den = 0.f;
        for (int k = 0; k < TOPK; k++) {
            float e = __expf(key2f(finKey[k]) - mx);
            wgt[k] = e;
            den += e;
        }
        sDen = den;
    }
    __syncthreads();
    float inv = 1.0f / sDen;

    // pick[m][d] = sum_k w_k * E_w[d, topi_k]
    for (int d = t; d < DEn; d += nthr) {
        const float* erow = Ew + (size_t)d * Vn;
        float acc = 0.f;
        for (int k = 0; k < TOPK; k++) acc += wgt[k] * inv * erow[finIdx[k]];
        pick[(size_t)m * DEn + d] = acc;
    }
}

// ---------------- K5: out_vec = pick[128x256] @ W_out^T + b_out, scatter into output ----------------
__global__ void k5_out(const float* __restrict__ pick, const float* __restrict__ Wout,
                       const float* __restrict__ bout, const int* __restrict__ pos,
                       float* __restrict__ out) {
    int wave = threadIdx.x >> 5, lane = threadIdx.x & 31;
    int mt = wave;
    int nt = blockIdx.x;                // 64 n-tiles of 16 over D=1024
    int halfsel = lane >> 4, l15 = lane & 15;
    int koff = 2 * halfsel;
    int ncol = nt * 16 + l15;
    const float* aptr = pick + (size_t)(mt * 16 + l15) * DEn;
    const float* bptr = Wout + (size_t)ncol * DEn;   // B[k][n] = W_out[n*DEn + k]
    v8f c = {};
    #pragma unroll 4
    for (int k = 0; k < DEn; k += 4) {
        v2f a = *(const v2f*)(aptr + k + koff);
        v2f b = *(const v2f*)(bptr + k + koff);
        c = __builtin_amdgcn_wmma_f32_16x16x4_f32(false, a, false, b, (short)0, c, false, false);
    }
    float bb = bout[ncol];
    for (int i = 0; i < 8; i++) {
        int mrow = mt * 16 + i + 8 * halfsel;
        int p = pos[mrow];
        out[(size_t)p * Dn + ncol] = c[i] + bb;
    }
}

extern "C" void kernel_launch(void* const* d_in, const int* in_sizes, int n_in,
                              void* d_out, int out_size, void* d_ws, size_t ws_size,
                              hipStream_t stream) {
    const int*   ids  = (const int*)d_in[0];
    const float* hs   = (const float*)d_in[1];
    const int*   mask = (const int*)d_in[2];
    const float* Win  = (const float*)d_in[3];
    const float* bin  = (const float*)d_in[4];
    const float* Ew   = (const float*)d_in[5];
    const float* Wout = (const float*)d_in[6];
    const float* bout = (const float*)d_in[7];
    float* out = (float*)d_out;

    char* ws = (char*)d_ws;
    int*      pos   = (int*)(ws);                                   // 128 ints
    float*    query = (float*)(ws + 4096);                          // 128x256 f32
    float*    pick  = (float*)(ws + 4096 + (size_t)NMENT * DEn * 4);// 128x256 f32
    unsigned* skey  = (unsigned*)(ws + (1 << 20));                  // 128 x 1e6 u32 keys (512 MB)
    int* topi = (int*)(out + (size_t)Bn * Sn * Dn);                 // topi tail of d_out

    k0_zero <<<8192, 256, 0, stream>>>((float4*)out, (Bn * Sn * Dn) / 4);
    k1_find <<<1, 1024, 0, stream>>>(ids, pos);
    k2_query<<<16, 256, 0, stream>>>(hs, pos, Win, bin, query);
    k3_score<<<Vn / (16 * NT_PER_BLK), 256, 0, stream>>>(query, Ew, mask, skey);
    k4_topk <<<NMENT, 256, 0, stream>>>(skey, Ew, pick, topi);
    k5_out  <<<64, 256, 0, stream>>>(pick, Wout, bout, pos, out);
}